// PerFilterMLPHead_55791625175439
// MI455X (gfx1250) — compile-verified
//
#include <hip/hip_runtime.h>
#include <hip/hip_bf16.h>

typedef __attribute__((ext_vector_type(2))) float v2f;
typedef __attribute__((ext_vector_type(4))) float v4f;
typedef __attribute__((ext_vector_type(8))) float v8f;

#define B_DIM 128
#define C_DIM 256
#define HW 3136          // 56*56
#define HW4 784          // HW / 4
#define CD 2048          // C_DIM * PER_FILTER_DIM
#define HID 256
#define NCLS 1000
#define NCLS_PAD 1008    // 63 tiles of 16

// ---------------------------------------------------------------------------
// Kernel 1: global average pool. One block per (b,c) plane (contiguous HW
// floats). Nontemporal ext-vector float4 loads (411 MB streamed once, NT
// temporal hint keeps it from displacing the GEMM operands), LDS reduction.
// ---------------------------------------------------------------------------
__global__ void pfmlp_pool(const float* __restrict__ x, float* __restrict__ pooled) {
    const int plane = blockIdx.x;                       // b*C + c
    const v4f* src = (const v4f*)(x + (size_t)plane * HW);
    float sum = 0.f;
    for (int i = threadIdx.x; i < HW4; i += 256) {
        v4f v = __builtin_nontemporal_load(&src[i]);
        sum += v.x + v.y + v.z + v.w;
    }
    __shared__ float red[256];
    red[threadIdx.x] = sum;
    __syncthreads();
    for (int s = 128; s > 0; s >>= 1) {
        if (threadIdx.x < s) red[threadIdx.x] += red[threadIdx.x + s];
        __syncthreads();
    }
    if (threadIdx.x == 0) pooled[plane] = red[0] * (1.0f / (float)HW);
}

// ---------------------------------------------------------------------------
// Kernel 2: feats = relu(pooled[:,:,None] * W1 + b1) flattened to [B, C*D].
// ---------------------------------------------------------------------------
__global__ void pfmlp_feats(const float* __restrict__ pooled,
                            const float* __restrict__ W1,
                            const float* __restrict__ b1,
                            float* __restrict__ concat) {
    const int idx = blockIdx.x * 256 + threadIdx.x;     // B*C*D = 262144
    const int cd = idx & (CD - 1);                      // c*8 + d
    const int b  = idx >> 11;
    const int c  = cd >> 3;
    float v = pooled[b * C_DIM + c] * W1[cd] + b1[cd];
    concat[idx] = v > 0.f ? v : 0.f;
}

// ---------------------------------------------------------------------------
// Kernel 3: pad W3 [1000,256] -> W3p [1008,256] (zero rows 1000..1007) so the
// GEMM inner loop needs no bounds checks.
// ---------------------------------------------------------------------------
__global__ void pfmlp_pad_w3(const float* __restrict__ W3, float* __restrict__ W3p) {
    const int idx = blockIdx.x * 256 + threadIdx.x;     // NCLS_PAD*HID = 258048
    const int n = idx >> 8;                             // row (k = idx & 255)
    W3p[idx] = (n < NCLS) ? W3[idx] : 0.f;
}

// ---------------------------------------------------------------------------
// FP32 WMMA GEMM:  D[m,n] = act( sum_k A[m,k] * W[n,k] + bias[n] )
//   A: [M,K] row-major; W: [Npad,K] row-major (B = W^T), ALL Npad rows
//   loadable (padded); D: [M,Nvalid].  One wave per 16x16 tile via
//   V_WMMA_F32_16X16X4_F32, branch-free inner loop (EXEC all-1s throughout).
// f32 A(16x4) frag: lanes 0-15 -> K=0,1 ; lanes 16-31 -> K=2,3 (M = lane&15).
// f32 B(4x16) frag: mirrored (N = lane&15).  C/D: VGPR r -> M = r + 8*half.
// ---------------------------------------------------------------------------
template <int RELU>
__global__ void pfmlp_wmma_gemm(const float* __restrict__ A,
                                const float* __restrict__ W,
                                const float* __restrict__ bias,
                                float* __restrict__ Dmat,
                                int M, int K, int Nvalid, int nTiles) {
    const int wavesPerBlock = blockDim.x >> 5;
    const int tile = blockIdx.x * wavesPerBlock + (threadIdx.x >> 5);
    const int totalTiles = (M >> 4) * nTiles;
    if (tile >= totalTiles) return;                     // wave-uniform exit

    const int lane = threadIdx.x & 31;
    const int half = lane >> 4;                         // K pair select: 0,1 vs 2,3
    const int r    = lane & 15;
    const int mt = tile / nTiles;
    const int nt = tile - mt * nTiles;

    const int m = (mt << 4) + r;                        // A row for this lane
    const int n = (nt << 4) + r;                        // W row / output col

    const v2f* arow = (const v2f*)(A + (size_t)m * K) + half;
    const v2f* brow = (const v2f*)(W + (size_t)n * K) + half;

    v8f acc = {};
    const int steps = K >> 2;                           // K/4 WMMAs
    #pragma unroll 4
    for (int s = 0; s < steps; ++s) {
        v2f a = arow[s * 2];
        v2f b = brow[s * 2];
        acc = __builtin_amdgcn_wmma_f32_16x16x4_f32(
            /*neg_a=*/false, a, /*neg_b=*/false, b,
            /*c_mod=*/(short)0, acc, /*reuse_a=*/false, /*reuse_b=*/false);
    }

    if (n < Nvalid) {                                   // epilogue-only predication
        const float bv = bias[n];
        #pragma unroll
        for (int v = 0; v < 8; ++v) {
            const int mm = (mt << 4) + v + (half << 3);
            float val = acc[v] + bv;
            if (RELU) val = val > 0.f ? val : 0.f;
            Dmat[(size_t)mm * Nvalid + n] = val;
        }
    }
}

// ---------------------------------------------------------------------------
// Launch
// ---------------------------------------------------------------------------
extern "C" void kernel_launch(void* const* d_in, const int* in_sizes, int n_in,
                              void* d_out, int out_size, void* d_ws, size_t ws_size,
                              hipStream_t stream) {
    const float* x  = (const float*)d_in[0];
    const float* W1 = (const float*)d_in[1];
    const float* b1 = (const float*)d_in[2];
    const float* W2 = (const float*)d_in[3];
    const float* b2 = (const float*)d_in[4];
    const float* W3 = (const float*)d_in[5];
    const float* b3 = (const float*)d_in[6];
    float* out = (float*)d_out;

    // Workspace (floats): pooled[B*C] | concat[B*CD] | h[B*HID] | W3p[1008*HID]
    float* pooled = (float*)d_ws;
    float* concat = pooled + B_DIM * C_DIM;
    float* h      = concat + B_DIM * CD;
    float* W3p    = h + B_DIM * HID;

    // 1) Global average pool.
    pfmlp_pool<<<B_DIM * C_DIM, 256, 0, stream>>>(x, pooled);

    // 2) Per-filter Linear(1->8) + ReLU.
    pfmlp_feats<<<(B_DIM * CD) / 256, 256, 0, stream>>>(pooled, W1, b1, concat);

    // 3) Zero-pad W3 to 1008 rows (no dependency on 1-2).
    pfmlp_pad_w3<<<(NCLS_PAD * HID) / 256, 256, 0, stream>>>(W3, W3p);

    // 4) h = relu(concat @ W2^T + b2)   M=128, K=2048, N=256 -> 128 tiles
    {
        const int nTiles = HID / 16;                    // 16
        const int blocks = (B_DIM / 16) * nTiles / 8;   // 128 tiles / 8 waves = 16
        pfmlp_wmma_gemm<1><<<blocks, 256, 0, stream>>>(concat, W2, b2, h,
                                                       B_DIM, CD, HID, nTiles);
    }

    // 5) out = h @ W3p^T + b3   M=128, K=256, Npad=1008 -> 504 tiles
    {
        const int nTiles = NCLS_PAD / 16;               // 63
        const int blocks = (B_DIM / 16) * nTiles / 8;   // 504 / 8 = 63
        pfmlp_wmma_gemm<0><<<blocks, 256, 0, stream>>>(h, W3p, b3, out,
                                                       B_DIM, HID, NCLS, nTiles);
    }
}